// KPConvResBlock_14817637171673
// MI455X (gfx1250) — compile-verified
//
#include <hip/hip_runtime.h>
#include <hip/hip_bf16.h>

// ---------------------------------------------------------------------------
// KPConv residual block for MI455X (gfx1250), fp32 WMMA path.
//   N=80000, H=32, K=15, CIN=128, COUT=256, D2=64
// v2: register-blocked GEMMs — each wave owns a 16x64 output strip with 4
// WMMA accumulators so one A fragment feeds 4 v_wmma_f32_16x16x4_f32 issues
// (4x fewer A fetches, ~4x better wmma:vmem ratio in the hot loops).
// ---------------------------------------------------------------------------

typedef __attribute__((ext_vector_type(2))) float v2f;
typedef __attribute__((ext_vector_type(8))) float v8f;

#define CIN   128
#define COUT  256
#define D2    64
#define KPTS  15
#define NBH   32
#define KDFLAT (KPTS * D2)          // 960
#define INFL_INV 2.5f               // 1/0.4
#define BN_EPS 1e-5f
#define SLOPE 0.2f

__device__ __forceinline__ v8f wmma_f32(v2f a, v2f b, v8f c) {
  return __builtin_amdgcn_wmma_f32_16x16x4_f32(
      /*neg_a=*/false, a, /*neg_b=*/false, b,
      /*c_mod=*/(short)0, c, /*reuse_a=*/false, /*reuse_b=*/false);
}

__device__ __forceinline__ float lrelu(float v) {
  return v >= 0.0f ? v : SLOPE * v;
}

// ---------------------------------------------------------------------------
// Row-major fp32 GEMM: C[M,Nc] = A[M,Kd] @ B[Kd,Nc].
// One wave per 16x64 output strip (4 accumulators); K-loop in steps of 4.
// Requires M%16==0, Kd%4==0, Nc%64==0.
// ---------------------------------------------------------------------------
__global__ __launch_bounds__(256) void gemm_f32_wmma(
    const float* __restrict__ A, const float* __restrict__ B,
    float* __restrict__ C, int M, int Kd, int Nc, int totalUnits) {
  int unit = blockIdx.x * (blockDim.x >> 5) + (threadIdx.x >> 5);
  if (unit >= totalUnits) return;                 // wave-uniform exit
  int lane = threadIdx.x & 31;
  int m    = lane & 15;
  int half = lane >> 4;                           // 0: K+0/K+1, 1: K+2/K+3
  int strips = Nc >> 6;                           // 64-wide column strips
  int row0 = (unit / strips) << 4;
  int col0 = (unit % strips) << 6;

  const float* Arow = A + (size_t)(row0 + m) * Kd;
  v8f acc0 = {}, acc1 = {}, acc2 = {}, acc3 = {};
  for (int kk = 0; kk < Kd; kk += 4) {
    int ka = kk + (half << 1);
    v2f a;                                        // A[m][ka], A[m][ka+1]
    a.x = Arow[ka];
    a.y = Arow[ka + 1];
    const float* Bp = B + (size_t)ka * Nc + col0 + m;
    v2f b0, b1, b2, b3;                           // B[ka..ka+1][col0+ct*16+m]
    b0.x = Bp[0];  b0.y = Bp[Nc];
    b1.x = Bp[16]; b1.y = Bp[Nc + 16];
    b2.x = Bp[32]; b2.y = Bp[Nc + 32];
    b3.x = Bp[48]; b3.y = Bp[Nc + 48];
    acc0 = wmma_f32(a, b0, acc0);
    acc1 = wmma_f32(a, b1, acc1);
    acc2 = wmma_f32(a, b2, acc2);
    acc3 = wmma_f32(a, b3, acc3);
  }
  // C/D layout: VGPR r -> row (row0 + half*8 + r), col (col0 + ct*16 + m)
  int rbase = row0 + (half << 3);
  float* Cp = C + (size_t)rbase * Nc + col0 + m;
  #pragma unroll
  for (int r = 0; r < 8; ++r) {
    size_t o = (size_t)r * Nc;
    Cp[o +  0] = acc0[r];
    Cp[o + 16] = acc1[r];
    Cp[o + 32] = acc2[r];
    Cp[o + 48] = acc3[r];
  }
}

// ---------------------------------------------------------------------------
// KPConv: one workgroup (256 thr = 8 waves) handles 16 points.
// Phase 1 (each wave: 2 points): influence matrix w[h][k] -> LDS, then
//   fk[16(k) x 64(d)] = w^T[16x32] @ x1[nbrs][32x64] via 32 WMMA/point
//   (one A frag -> 4 WMMAs), stored to LDS FK[point][k*64+d].
// Phase 2 (8 waves, each a 120-deep K-slice of the full 16x64 strip):
//   xk[16x64] = FK[16x960] @ kpw_flat[960x64], 120 WMMA/wave, partials in LDS.
// Dynamic LDS: FK 61440B + Wl 16384B + IDX 1024B + PS 32768B = 111616B.
// ---------------------------------------------------------------------------
__global__ __launch_bounds__(256) void kpconv_wmma(
    const float* __restrict__ xyz, const int* __restrict__ nidx,
    const float* __restrict__ x1, const float* __restrict__ kp_points,
    const float* __restrict__ kpw, float* __restrict__ xk) {
  extern __shared__ float smem[];
  float* FK  = smem;                         // [16][960]
  float* Wl  = FK + 16 * KDFLAT;             // [8 waves][32 h][16 k]
  int*   IDX = (int*)(Wl + 8 * NBH * 16);    // [8 waves][32]
  float* PS  = (float*)(IDX + 8 * NBH);      // [8][16][64] partials

  const int wave = threadIdx.x >> 5;
  const int lane = threadIdx.x & 31;
  const int m    = lane & 15;
  const int half = lane >> 4;
  const int base = blockIdx.x << 4;          // first point of this block

  float* wl = Wl + wave * (NBH * 16);
  int*   il = IDX + wave * NBH;

  // ---------------- Phase 1: build FK for 16 points ----------------
  for (int it = 0; it < 2; ++it) {
    const int pi = (wave << 1) + it;         // 0..15 within block
    const int n  = base + pi;

    // lane = neighbor h
    const int id = nidx[(size_t)n * NBH + lane];
    il[lane] = id;
    const float cx = xyz[(size_t)n * 3 + 0];
    const float cy = xyz[(size_t)n * 3 + 1];
    const float cz = xyz[(size_t)n * 3 + 2];
    const float px = xyz[(size_t)id * 3 + 0] - cx;
    const float py = xyz[(size_t)id * 3 + 1] - cy;
    const float pz = xyz[(size_t)id * 3 + 2] - cz;
    #pragma unroll
    for (int k = 0; k < KPTS; ++k) {
      float dx = px - kp_points[k * 3 + 0];
      float dy = py - kp_points[k * 3 + 1];
      float dz = pz - kp_points[k * 3 + 2];
      float r = sqrtf(dx * dx + dy * dy + dz * dz);
      wl[lane * 16 + k] = fmaxf(1.0f - r * INFL_INV, 0.0f);
    }
    wl[lane * 16 + KPTS] = 0.0f;             // pad k=15 row with zeros
    __syncthreads();                         // uniform: all waves loop twice

    // fk = w^T @ nx : 16x64 strip, K-depth 32 (neighbors), 4 acc
    v8f acc0 = {}, acc1 = {}, acc2 = {}, acc3 = {};
    #pragma unroll
    for (int kk = 0; kk < NBH; kk += 4) {
      const int ka = kk + (half << 1);       // neighbor index h
      v2f a;                                 // A[k=m][h=ka] = w[ka][m]
      a.x = wl[ka * 16 + m];
      a.y = wl[(ka + 1) * 16 + m];
      const float* r0 = x1 + (size_t)il[ka] * D2 + m;
      const float* r1 = x1 + (size_t)il[ka + 1] * D2 + m;
      v2f b0, b1, b2, b3;                    // B[h][d] = x1[nbr][d]
      b0.x = r0[0];  b0.y = r1[0];
      b1.x = r0[16]; b1.y = r1[16];
      b2.x = r0[32]; b2.y = r1[32];
      b3.x = r0[48]; b3.y = r1[48];
      acc0 = wmma_f32(a, b0, acc0);
      acc1 = wmma_f32(a, b1, acc1);
      acc2 = wmma_f32(a, b2, acc2);
      acc3 = wmma_f32(a, b3, acc3);
    }
    // store rows k=0..14 (row 15 is zero by construction)
    #pragma unroll
    for (int r = 0; r < 8; ++r) {
      const int krow = (half << 3) + r;
      if (krow < KPTS) {
        float* f = FK + pi * KDFLAT + krow * D2 + m;
        f[0]  = acc0[r];
        f[16] = acc1[r];
        f[32] = acc2[r];
        f[48] = acc3[r];
      }
    }
    __syncthreads();
  }

  // ---------------- Phase 2: xk = FK @ kpw_flat ----------------
  // wave w handles K-slice [w*120, w*120+120) of the full 16x64 strip
  {
    const int kbeg = wave * 120;
    v8f acc0 = {}, acc1 = {}, acc2 = {}, acc3 = {};
    for (int kk = kbeg; kk < kbeg + 120; kk += 4) {
      const int ka = kk + (half << 1);       // flat index k*64+d
      v2f a;                                 // A[point=m][ka]
      a.x = FK[m * KDFLAT + ka];
      a.y = FK[m * KDFLAT + ka + 1];
      const float* Bp = kpw + (size_t)ka * D2 + m;  // B[ka][e] (flat kpw)
      v2f b0, b1, b2, b3;
      b0.x = Bp[0];  b0.y = Bp[D2];
      b1.x = Bp[16]; b1.y = Bp[D2 + 16];
      b2.x = Bp[32]; b2.y = Bp[D2 + 32];
      b3.x = Bp[48]; b3.y = Bp[D2 + 48];
      acc0 = wmma_f32(a, b0, acc0);
      acc1 = wmma_f32(a, b1, acc1);
      acc2 = wmma_f32(a, b2, acc2);
      acc3 = wmma_f32(a, b3, acc3);
    }
    #pragma unroll
    for (int r = 0; r < 8; ++r) {
      float* p = PS + wave * (16 * D2) + ((half << 3) + r) * D2 + m;
      p[0]  = acc0[r];
      p[16] = acc1[r];
      p[32] = acc2[r];
      p[48] = acc3[r];
    }
  }
  __syncthreads();

  // reduce the 8 K-slice partials and write out
  for (int t = threadIdx.x; t < 16 * D2; t += 256) {
    float s = 0.0f;
    #pragma unroll
    for (int w = 0; w < 8; ++w) s += PS[w * (16 * D2) + t];
    const int pt = t >> 6;
    const int e  = t & 63;
    xk[(size_t)(base + pt) * D2 + e] = s;
  }
}

// ---------------------------------------------------------------------------
// BN helpers
// ---------------------------------------------------------------------------
__global__ void zero_f32(float* __restrict__ p, int n) {
  int i = blockIdx.x * blockDim.x + threadIdx.x;
  if (i < n) p[i] = 0.0f;
}

// per-column sum & sumsq of row-major Y[n][C]; stats = [sum[C], sumsq[C]]
__global__ void colstats(const float* __restrict__ Y, long long total, int C,
                         float* __restrict__ stats) {
  size_t idx0   = (size_t)blockIdx.x * blockDim.x + threadIdx.x;
  size_t stride = (size_t)gridDim.x * blockDim.x;  // multiple of C (C | 256)
  float s = 0.0f, s2 = 0.0f;
  for (size_t i = idx0; i < (size_t)total; i += stride) {
    float v = Y[i];
    s += v;
    s2 += v * v;
  }
  int col = (int)(idx0 % (size_t)C);
  atomicAdd(&stats[col], s);
  atomicAdd(&stats[C + col], s2);
}

__global__ void finalize_bn(const float* __restrict__ stats,
                            const float* __restrict__ g,
                            const float* __restrict__ b, int C, float invN,
                            float* __restrict__ sc, float* __restrict__ sh) {
  int c = blockIdx.x * blockDim.x + threadIdx.x;
  if (c < C) {
    float mean = stats[c] * invN;
    float var  = stats[C + c] * invN - mean * mean;  // biased, matches jnp.var
    float s    = g[c] * rsqrtf(var + BN_EPS);
    sc[c] = s;
    sh[c] = b[c] - mean * s;
  }
}

// in-place BN + leaky-relu on row-major [n][C]
__global__ void bnact_inplace(float* __restrict__ Y, const float* __restrict__ sc,
                              const float* __restrict__ sh, long long total, int C) {
  size_t idx0   = (size_t)blockIdx.x * blockDim.x + threadIdx.x;
  size_t stride = (size_t)gridDim.x * blockDim.x;
  for (size_t i = idx0; i < (size_t)total; i += stride) {
    int c = (int)(i % (size_t)C);
    Y[i] = lrelu(Y[i] * sc[c] + sh[c]);
  }
}

// out = lrelu(bn2(out)) + bn_sc(ysc)
__global__ void fuse_out(float* __restrict__ out, const float* __restrict__ ysc,
                         const float* __restrict__ sc2, const float* __restrict__ sh2,
                         const float* __restrict__ scS, const float* __restrict__ shS,
                         long long total, int C) {
  size_t idx0   = (size_t)blockIdx.x * blockDim.x + threadIdx.x;
  size_t stride = (size_t)gridDim.x * blockDim.x;
  for (size_t i = idx0; i < (size_t)total; i += stride) {
    int c = (int)(i % (size_t)C);
    float v = lrelu(out[i] * sc2[c] + sh2[c]);
    out[i] = v + (ysc[i] * scS[c] + shS[c]);
  }
}

// ---------------------------------------------------------------------------
extern "C" void kernel_launch(void* const* d_in, const int* in_sizes, int n_in,
                              void* d_out, int out_size, void* d_ws, size_t ws_size,
                              hipStream_t stream) {
  const float* feats = (const float*)d_in[0];
  const float* xyz   = (const float*)d_in[1];
  const int*   nidx  = (const int*)d_in[3];
  const float* W1    = (const float*)d_in[4];
  const float* g1    = (const float*)d_in[5];
  const float* b1    = (const float*)d_in[6];
  const float* kpp   = (const float*)d_in[7];
  const float* kpw   = (const float*)d_in[8];
  const float* W2    = (const float*)d_in[9];
  const float* g2    = (const float*)d_in[10];
  const float* b2    = (const float*)d_in[11];
  const float* Wsc   = (const float*)d_in[12];
  const float* gsc   = (const float*)d_in[13];
  const float* bsc   = (const float*)d_in[14];
  float* out = (float*)d_out;

  const int N = in_sizes[0] / CIN;              // 80000 (multiple of 16)

  // workspace layout (floats)
  float* ws      = (float*)d_ws;
  float* x1      = ws;                          // [N, D2]
  float* xk      = x1 + (size_t)N * D2;         // [N, D2]
  float* ysc     = xk + (size_t)N * D2;         // [N, COUT]
  float* stats1  = ysc + (size_t)N * COUT;      // 2*D2
  float* stats2  = stats1 + 2 * D2;             // 2*COUT
  float* statsSC = stats2 + 2 * COUT;           // 2*COUT
  float* sc1 = statsSC + 2 * COUT; float* sh1 = sc1 + D2;
  float* sc2 = sh1 + D2;           float* sh2 = sc2 + COUT;
  float* scS = sh2 + COUT;         float* shS = scS + COUT;

  const int statN = 2 * D2 + 4 * COUT;          // 1152
  zero_f32<<<(statN + 255) / 256, 256, 0, stream>>>(stats1, statN);

  // ---- unary_1 GEMM: feats[N,128] @ W1[128,64] -> x1 (pre-BN) ----
  {
    int units = (N / 16) * (D2 / 64);
    gemm_f32_wmma<<<(units + 7) / 8, 256, 0, stream>>>(feats, W1, x1, N, CIN, D2, units);
  }
  colstats<<<256, 256, 0, stream>>>(x1, (long long)N * D2, D2, stats1);
  finalize_bn<<<1, D2, 0, stream>>>(stats1, g1, b1, D2, 1.0f / (float)N, sc1, sh1);
  bnact_inplace<<<2048, 256, 0, stream>>>(x1, sc1, sh1, (long long)N * D2, D2);

  // ---- KPConv -> xk [N, D2] ----
  {
    size_t lds = (16 * KDFLAT + 8 * NBH * 16 + 8 * 16 * D2) * sizeof(float)
               + 8 * NBH * sizeof(int);        // 111616 bytes
    kpconv_wmma<<<N / 16, 256, lds, stream>>>(xyz, nidx, x1, kpp, kpw, xk);
  }

  // ---- unary_2 GEMM: xk[N,64] @ W2[64,256] -> out (pre-BN y2) ----
  {
    int units = (N / 16) * (COUT / 64);
    gemm_f32_wmma<<<(units + 7) / 8, 256, 0, stream>>>(xk, W2, out, N, D2, COUT, units);
  }
  // ---- shortcut GEMM: feats[N,128] @ Wsc[128,256] -> ysc (pre-BN) ----
  {
    int units = (N / 16) * (COUT / 64);
    gemm_f32_wmma<<<(units + 7) / 8, 256, 0, stream>>>(feats, Wsc, ysc, N, CIN, COUT, units);
  }

  colstats<<<256, 256, 0, stream>>>(out, (long long)N * COUT, COUT, stats2);
  colstats<<<256, 256, 0, stream>>>(ysc, (long long)N * COUT, COUT, statsSC);
  finalize_bn<<<1, COUT, 0, stream>>>(stats2, g2, b2, COUT, 1.0f / (float)N, sc2, sh2);
  finalize_bn<<<1, COUT, 0, stream>>>(statsSC, gsc, bsc, COUT, 1.0f / (float)N, scS, shS);

  // ---- out = lrelu(bn2(y2)) + bn_sc(ysc) ----
  fuse_out<<<4096, 256, 0, stream>>>(out, ysc, sc2, sh2, scS, shS,
                                     (long long)N * COUT, COUT);
}